// geo_loss_v3_9062380995247
// MI455X (gfx1250) — compile-verified
//
#include <hip/hip_runtime.h>
#include <math.h>

// Problem constants (match reference: B=2, Nt=2048, UP_RATIO=10, K=5)
#define BB 2
#define NT 2048
#define UPR 10
#define QN (NT * (UPR + 1))   // 22528 queries per batch
#define QTILES (QN / 32)      // 704 waves of 32 queries per batch
#define ROWS 20               // LDS row stride in words (16B-aligned rows)
#define IDXMASK 0x7FFu        // 11-bit point index (NT = 2048)
#define DMASK 0xFFFFF800u     // top-21 bits of monotonic distance

typedef float v2f __attribute__((ext_vector_type(2)));
typedef float v8f __attribute__((ext_vector_type(8)));

// Monotonic float->uint32 bijection: preserves IEEE ordering as unsigned.
__device__ __forceinline__ unsigned int f2mono(float f) {
  unsigned int u = __float_as_uint(f);
  return u ^ (0x80000000u | (unsigned int)((int)u >> 31));
}

// ---------------------------------------------------------------------------
// Kernel A: self-NN std + noisy query generation.
// Whole 2048-point cloud staged in 24KB LDS; each thread scans for its NN.
// ---------------------------------------------------------------------------
__global__ __launch_bounds__(256) void std_query_kernel(
    const float* __restrict__ tgt, const float* __restrict__ noise,
    float* __restrict__ query) {
  __shared__ float sp[NT * 3];
  const int b = blockIdx.y;
  const float* tb = tgt + (size_t)b * NT * 3;
  for (int t = threadIdx.x; t < NT * 3; t += 256) sp[t] = tb[t];
  __syncthreads();

  const int i = blockIdx.x * 256 + threadIdx.x;
  const float x = sp[i * 3 + 0], y = sp[i * 3 + 1], z = sp[i * 3 + 2];
  float best = 3.402823466e38f;
  for (int j = 0; j < NT; ++j) {
    float dx = x - sp[j * 3 + 0];
    float dy = y - sp[j * 3 + 1];
    float dz = z - sp[j * 3 + 2];
    float d = fmaf(dx, dx, fmaf(dy, dy, dz * dz));
    best = (j == i) ? best : fminf(best, d);  // skip self (dist-0 slot)
  }
  const float stdv = sqrtf(best + 1e-10f) * 10.0f;  // STD_FACTOR

  float* qb = query + (size_t)b * QN * 3;
  const float* nb = noise + ((size_t)b * NT + i) * UPR * 3;
#pragma unroll
  for (int u = 0; u < UPR; ++u) {
    const int qi = (i * UPR + u) * 3;
    qb[qi + 0] = fmaf(nb[u * 3 + 0], stdv, x);
    qb[qi + 1] = fmaf(nb[u * 3 + 1], stdv, y);
    qb[qi + 2] = fmaf(nb[u * 3 + 2], stdv, z);
  }
}

// ---------------------------------------------------------------------------
// Kernel B: append src points to the query array tail (concat along Q axis).
// ---------------------------------------------------------------------------
__global__ __launch_bounds__(256) void copy_src_kernel(
    const float* __restrict__ src, float* __restrict__ query) {
  const int idx = blockIdx.x * 256 + threadIdx.x;
  if (idx >= BB * NT * 3) return;
  const int b = idx / (NT * 3);
  const int r = idx % (NT * 3);
  query[(size_t)b * QN * 3 + NT * UPR * 3 + r] = src[idx];
}

// ---------------------------------------------------------------------------
// WMMA brute-force top-5 KNN of 32 queries (two 16x4 f32 A-fragments) vs 2048
// points. Per tile: V_WMMA_F32_16X16X4_F32 gives q.p; d' = |p|^2 - 2 q.p is
// mapped to a monotonic u32, truncated to 21 bits and packed with the column
// index before the LDS exchange. Each lane scans its own query's row with 4x
// ds_load_b128; a candidate key is just (stored + tile_base) and the top-5 is
// kept by a branchless 32-bit compare/swap bubble (no exec divergence, ties
// broken stably by index). Exact distances are recomputed in the epilogue.
// ---------------------------------------------------------------------------
__device__ __forceinline__ void knn_top5(const float* __restrict__ pts,
                                         v2f a0, v2f a1, int lane,
                                         unsigned int* tile,
                                         unsigned int (&bk)[5]) {
#pragma unroll
  for (int i = 0; i < 5; ++i) bk[i] = 0xFFFFFFFFu;
  const unsigned int n = (unsigned int)(lane & 15);
  const int hi8 = (lane >> 4) << 3;

  for (int t = 0; t < NT / 16; ++t) {
    // B fragment: 4x16 points, lanes 0-15 hold (x,y), lanes 16-31 hold (z,0)
    const float* pp = pts + (t * 16 + n) * 3;
    const float px = pp[0], py = pp[1], pz = pp[2];
    const float pn = fmaf(px, px, fmaf(py, py, pz * pz));
    v2f bf;
    bf.x = (lane >= 16) ? pz : px;
    bf.y = (lane >= 16) ? 0.0f : py;

    v8f cz = {};
    v8f d0 = __builtin_amdgcn_wmma_f32_16x16x4_f32(
        false, a0, false, bf, (short)0, cz, false, false);
    v8f d1 = __builtin_amdgcn_wmma_f32_16x16x4_f32(
        false, a1, false, bf, (short)0, cz, false, false);

#pragma unroll
    for (int v = 0; v < 8; ++v) {
      tile[(v + hi8) * ROWS + n] =
          (f2mono(fmaf(-2.0f, d0[v], pn)) & DMASK) | n;
      tile[(16 + v + hi8) * ROWS + n] =
          (f2mono(fmaf(-2.0f, d1[v], pn)) & DMASK) | n;
    }
    __syncthreads();  // single-wave WG: barrier is free, orders DS ops

    const unsigned int base = (unsigned int)(t * 16);  // fits in 11-bit field
    const uint4* row = (const uint4*)(tile + lane * ROWS);
#pragma unroll
    for (int k = 0; k < 4; ++k) {
      const uint4 q4 = row[k];  // ds_load_b128
      const unsigned int mv[4] = {q4.x, q4.y, q4.z, q4.w};
#pragma unroll
      for (int e = 0; e < 4; ++e) {
        unsigned int c = mv[e] + base;  // key = trunc-dist | point index
#pragma unroll
        for (int s = 0; s < 5; ++s) {  // branchless sorted insert (u32)
          const bool lt = c < bk[s];
          const unsigned int lo = lt ? c : bk[s];
          const unsigned int hi = lt ? bk[s] : c;
          bk[s] = lo;
          c = hi;
        }
      }
    }
    __syncthreads();
  }
}

// ---------------------------------------------------------------------------
// Kernel C: one wave per 32 queries. KNN vs tgt and src, exact distance +
// direction recompute from gathered neighbors, softmax weights, UDF value +
// gradient errors, weighted scalar loss -> atomic accumulate.
// ---------------------------------------------------------------------------
__global__ __launch_bounds__(32) void udf_loss_kernel(
    const float* __restrict__ src, const float* __restrict__ tgt,
    const float* __restrict__ query, float* __restrict__ out) {
  __shared__ unsigned int tile[32 * ROWS];
  const int lane = threadIdx.x;
  const int b = blockIdx.x / QTILES;
  const int qbase = (blockIdx.x % QTILES) * 32;
  const float* qb = query + (size_t)b * QN * 3;

  // this lane's query (for the epilogue)
  const float qx = qb[(qbase + lane) * 3 + 0];
  const float qy = qb[(qbase + lane) * 3 + 1];
  const float qz = qb[(qbase + lane) * 3 + 2];

  // A fragments: 16x4 (MxK); lanes 0-15 hold K=(0,1)=(x,y), 16-31 K=(2,3)=(z,0)
  const int m = lane & 15;
  const int hi = lane >> 4;
  const float* q0 = qb + (qbase + m) * 3;
  const float* q1 = qb + (qbase + 16 + m) * 3;
  v2f a0, a1;
  a0.x = hi ? q0[2] : q0[0];
  a0.y = hi ? 0.0f : q0[1];
  a1.x = hi ? q1[2] : q1[0];
  a1.y = hi ? 0.0f : q1[1];

  const float* tp = tgt + (size_t)b * NT * 3;
  const float* sp = src + (size_t)b * NT * 3;
  unsigned int kt[5], ks[5];
  knn_top5(tp, a0, a1, lane, tile, kt);
  knn_top5(sp, a0, a1, lane, tile, ks);

  // Exact distances + directions to the 5 tgt neighbors (loads feed both
  // the distance and the gradient term).
  float dt[5], dirt[5][3];
#pragma unroll
  for (int i = 0; i < 5; ++i) {
    const float* nn = tp + (kt[i] & IDXMASK) * 3;
    const float dx = qx - nn[0], dy = qy - nn[1], dz = qz - nn[2];
    dirt[i][0] = dx; dirt[i][1] = dy; dirt[i][2] = dz;
    dt[i] = fmaf(dx, dx, fmaf(dy, dy, dz * dz));  // >= 0 by construction
  }

  // weights = softmax(-d_tgt) over K (shift by min for stability; exact)
  float dmin = dt[0];
#pragma unroll
  for (int i = 1; i < 5; ++i) dmin = fminf(dmin, dt[i]);
  float w[5], wsum = 0.0f;
#pragma unroll
  for (int i = 0; i < 5; ++i) {
    w[i] = __expf(dmin - dt[i]);
    wsum += w[i];
  }
  const float winv = 1.0f / wsum;

  float udf_t = 0.0f, gtx = 0.0f, gty = 0.0f, gtz = 0.0f;
#pragma unroll
  for (int i = 0; i < 5; ++i) {
    const float wi = w[i] * winv;
    udf_t += sqrtf(dt[i] + 1e-10f) * wi;
    gtx = fmaf(dirt[i][0], wi, gtx);
    gty = fmaf(dirt[i][1], wi, gty);
    gtz = fmaf(dirt[i][2], wi, gtz);
  }

  float udf_s = 0.0f, gsx = 0.0f, gsy = 0.0f, gsz = 0.0f;
#pragma unroll
  for (int i = 0; i < 5; ++i) {
    const float wi = w[i] * winv;  // same weights (from tgt dists) per reference
    const float* nn = sp + (ks[i] & IDXMASK) * 3;
    const float dx = qx - nn[0], dy = qy - nn[1], dz = qz - nn[2];
    const float ds2 = fmaf(dx, dx, fmaf(dy, dy, dz * dz));
    udf_s += sqrtf(ds2 + 1e-10f) * wi;
    gsx = fmaf(dx, wi, gsx);
    gsy = fmaf(dy, wi, gsy);
    gsz = fmaf(dz, wi, gsz);
  }

  const float err = fabsf(udf_t - udf_s);
  const float gerr = fabsf(gsx - gtx) + fabsf(gsy - gty) + fabsf(gsz - gtz);
  const float tot = err + gerr;
  float c = tot * __expf(-3.0f * tot) * (1.0f / ((float)BB * (float)QN));

#pragma unroll
  for (int off = 16; off > 0; off >>= 1) c += __shfl_xor(c, off);
  if (lane == 0) atomicAdd(out, c);
}

// ---------------------------------------------------------------------------
// Launch: inputs are src[2,2048,3], tgt[2,2048,3], noise[2,2048,10,3] (f32).
// Workspace holds the query array: B*QN*3 floats = 540,672 bytes.
// ---------------------------------------------------------------------------
extern "C" void kernel_launch(void* const* d_in, const int* in_sizes, int n_in,
                              void* d_out, int out_size, void* d_ws,
                              size_t ws_size, hipStream_t stream) {
  const float* src = (const float*)d_in[0];
  const float* tgt = (const float*)d_in[1];
  const float* noise = (const float*)d_in[2];
  float* out = (float*)d_out;
  float* query = (float*)d_ws;

  hipMemsetAsync(out, 0, sizeof(float), stream);  // graph-capturable
  std_query_kernel<<<dim3(NT / 256, BB), 256, 0, stream>>>(tgt, noise, query);
  copy_src_kernel<<<(BB * NT * 3 + 255) / 256, 256, 0, stream>>>(src, query);
  udf_loss_kernel<<<BB * QTILES, 32, 0, stream>>>(src, tgt, query, out);
}